// KNNGraph_81484119540282
// MI455X (gfx1250) — compile-verified
//
#include <hip/hip_runtime.h>
#include <hip/hip_bf16.h>

typedef float v2f __attribute__((ext_vector_type(2)));
typedef float v8f __attribute__((ext_vector_type(8)));

#define M_PTS 8192
#define D_DIM 64
#define K_NN  16
#define ROWS_PER_BLK 32
#define FLT_BIG 3.402823e38f

// One wave32 per block; block owns 32 query rows of one batch (two 16-row
// WMMA tiles sharing each B fragment). grid.x = N * (M_PTS/32).
__global__ __launch_bounds__(32) void knn_wmma_kernel(const float* __restrict__ x,
                                                      float* __restrict__ out_src,
                                                      float* __restrict__ out_dst)
{
    __shared__ float x2row_s[ROWS_PER_BLK];
    __shared__ float dtile[ROWS_PER_BLK][16];
    __shared__ float best_d[ROWS_PER_BLK][K_NN];
    __shared__ int   best_i[ROWS_PER_BLK][K_NN];

    const int lane    = threadIdx.x;              // 0..31
    const int bid     = blockIdx.x;
    const int n       = bid >> 8;                 // / (M_PTS/32)
    const int rowbase = (bid & 255) << 5;

    const int half = lane >> 4;                   // half-wave id
    const int l16  = lane & 15;

    // ---- Load A fragments for both 16-row tiles (kept in VGPRs) ----
    // ISA 32-bit A 16x4 layout: lane L -> M = L%16; K pair {0,1} (lanes<16) / {2,3}.
    const float* arow0 = x + ((size_t)(n * M_PTS + rowbase + l16)) * D_DIM + half * 2;
    const float* arow1 = arow0 + 16 * D_DIM;
    v2f a0[16], a1[16];
    float s0 = 0.0f, s1 = 0.0f;
#pragma unroll
    for (int s = 0; s < 16; ++s) {
        float2 t0 = *reinterpret_cast<const float2*>(arow0 + 4 * s);
        float2 t1 = *reinterpret_cast<const float2*>(arow1 + 4 * s);
        a0[s].x = t0.x; a0[s].y = t0.y;
        a1[s].x = t1.x; a1[s].y = t1.y;
        s0 = fmaf(t0.x, t0.x, fmaf(t0.y, t0.y, s0));
        s1 = fmaf(t1.x, t1.x, fmaf(t1.y, t1.y, s1));
    }
    const float full0 = s0 + __shfl_xor(s0, 16, 32);   // |x|^2 of row rowbase+l16
    const float full1 = s1 + __shfl_xor(s1, 16, 32);   // |x|^2 of row rowbase+16+l16
    if (lane < 16) { x2row_s[l16] = full0; x2row_s[16 + l16] = full1; }
    __syncthreads();

    // Hoist row norms into registers laid out to match the C-matrix mapping:
    // acc VGPR r, lanes 0-15 -> M=r, lanes 16-31 -> M=r+8.
    float x2r0[8], x2r1[8];
#pragma unroll
    for (int r = 0; r < 8; ++r) {
        x2r0[r] = x2row_s[r + half * 8];
        x2r1[r] = x2row_s[16 + r + half * 8];
    }

    // ---- Init per-row top-k lists; every lane owns row == lane ----
#pragma unroll
    for (int t = 0; t < K_NN; ++t) { best_d[lane][t] = FLT_BIG; best_i[lane][t] = 0; }
    __syncthreads();

    float cur_max  = FLT_BIG;   // lane's row 16th-best distance
    int   max_slot = 0;
    float wave_thr = FLT_BIG;   // uniform: max over rows of cur_max

    for (int ct = 0; ct < M_PTS / 16; ++ct) {
        const int colbase = ct << 4;

        // ---- B fragments for this column tile (shared by both row tiles) ----
        const float* brow = x + ((size_t)(n * M_PTS + colbase + l16)) * D_DIM + half * 2;
        v2f bfrag[16];
        float c2h = 0.0f;
#pragma unroll
        for (int s = 0; s < 16; ++s) {
            float2 t = *reinterpret_cast<const float2*>(brow + 4 * s);
            bfrag[s].x = t.x; bfrag[s].y = t.y;
            c2h = fmaf(t.x, t.x, fmaf(t.y, t.y, c2h));
        }
        const float x2col = c2h + __shfl_xor(c2h, 16, 32); // |col|^2, col = colbase+l16

        // ---- Two independent 16-WMMA chains (32 WMMAs / 16 B loads) ----
        v8f acc0 = {};
        v8f acc1 = {};
#pragma unroll
        for (int s = 0; s < 16; ++s) {
            acc0 = __builtin_amdgcn_wmma_f32_16x16x4_f32(
                false, a0[s], false, bfrag[s], (short)0, acc0, false, false);
            acc1 = __builtin_amdgcn_wmma_f32_16x16x4_f32(
                false, a1[s], false, bfrag[s], (short)0, acc1, false, false);
        }

        // ---- Distances in registers ----
        float d0[8], d1[8];
#pragma unroll
        for (int r = 0; r < 8; ++r) {
            d0[r] = fmaf(-2.0f, acc0[r], x2r0[r] + x2col);
            d1[r] = fmaf(-2.0f, acc1[r], x2r1[r] + x2col);
        }

        // ---- Fast rejection: can any of this tile's 512 distances enter a list? ----
        float mn = d0[0];
#pragma unroll
        for (int r = 1; r < 8; ++r) mn = fminf(mn, d0[r]);
#pragma unroll
        for (int r = 0; r < 8; ++r) mn = fminf(mn, d1[r]);
        // wave_thr >= every row's cutoff, so (mn >= wave_thr for all lanes) => safe skip.
        if (__ballot(mn < wave_thr) != 0ULL) {
            // Scatter the 32x16 distance tile.
#pragma unroll
            for (int r = 0; r < 8; ++r) {
                dtile[r + half * 8][l16]      = d0[r];
                dtile[16 + r + half * 8][l16] = d1[r];
            }
            __syncthreads();

            // Replace-max insertion; lane owns row == lane. Strict '<' keeps
            // the earlier (smaller) index on distance ties.
            for (int c = 0; c < 16; ++c) {
                const float d = dtile[lane][c];
                if (d < cur_max) {
                    best_d[lane][max_slot] = d;
                    best_i[lane][max_slot] = colbase + c;
                    float m = best_d[lane][0];
                    int   ms = 0;
#pragma unroll
                    for (int t = 1; t < K_NN; ++t) {
                        const float v = best_d[lane][t];
                        if (v > m) { m = v; ms = t; }
                    }
                    cur_max = m; max_slot = ms;
                }
            }
            __syncthreads();

            // Refresh the uniform threshold: wave-wide max of cur_max.
            float t = cur_max;
            t = fmaxf(t, __shfl_xor(t, 1,  32));
            t = fmaxf(t, __shfl_xor(t, 2,  32));
            t = fmaxf(t, __shfl_xor(t, 4,  32));
            t = fmaxf(t, __shfl_xor(t, 8,  32));
            t = fmaxf(t, __shfl_xor(t, 16, 32));
            wave_thr = t;
        }
    }

    // ---- Final per-row sort by (dist, idx) ascending; emit edges ----
    {
        const int row = lane;
        for (int i = 1; i < K_NN; ++i) {
            const float d  = best_d[row][i];
            const int   ix = best_i[row][i];
            int j = i - 1;
            while (j >= 0 && (best_d[row][j] > d ||
                              (best_d[row][j] == d && best_i[row][j] > ix))) {
                best_d[row][j + 1] = best_d[row][j];
                best_i[row][j + 1] = best_i[row][j];
                --j;
            }
            best_d[row][j + 1] = d;
            best_i[row][j + 1] = ix;
        }
        const int g = n * M_PTS + rowbase + row;
#pragma unroll
        for (int t = 0; t < K_NN; ++t) {
            out_src[(size_t)g * K_NN + t] = (float)(best_i[row][t] + n * M_PTS);
            out_dst[(size_t)g * K_NN + t] = (float)g;
        }
    }
}

extern "C" void kernel_launch(void* const* d_in, const int* in_sizes, int n_in,
                              void* d_out, int out_size, void* d_ws, size_t ws_size,
                              hipStream_t stream) {
    const float* x = (const float*)d_in[0];
    // d_in[1] is k (scalar int in device memory) -- fixed to 16 per the reference.
    const int total = in_sizes[0];                  // N * M * D
    const int N     = total / (M_PTS * D_DIM);      // = 4

    float* out_src = (float*)d_out;
    float* out_dst = out_src + (size_t)N * M_PTS * K_NN;

    dim3 grid(N * (M_PTS / ROWS_PER_BLK));
    dim3 block(32);
    knn_wmma_kernel<<<grid, block, 0, stream>>>(x, out_src, out_dst);
}